// MLE_27376121544766
// MI455X (gfx1250) — compile-verified
//
#include <hip/hip_runtime.h>
#include <hip/hip_bf16.h>

typedef __attribute__((ext_vector_type(2))) float v2f;
typedef __attribute__((ext_vector_type(8))) float v8f;

#define EPS          1e-5f
#define INV_SQRT2    0.70710678118654752f
#define HALF_LOG_2PI 0.91893853320467274f

// Per-sample log-normal survival loss, both branches computed + select
// (is_alive is random 0/1 -> waves diverge anyway; branchless keeps EXEC full).
__device__ __forceinline__ float per_sample_loss(float mu, float ls, float t, int alive) {
    // sigma = exp(ls); we only ever need 1/sigma and log(sigma)=ls,
    // so compute inv_sigma = exp(-ls) and skip the divides + one log.
    float inv_sigma = __expf(-ls);

    // alive branch: -log(1 - Phi(z) + eps)
    float z   = (__logf(t) - mu) * inv_sigma * INV_SQRT2;
    float cdf = 0.5f * (1.0f + erff(z));            // accurate erf to mimic reference
    float loss_alive = -__logf(1.0f - cdf + EPS);

    // dead branch: -log_prob
    float lx = __logf(t + EPS);
    float d  = lx - mu;
    float log_prob = -lx - ls - HALF_LOG_2PI - 0.5f * d * d * inv_sigma * inv_sigma;
    float loss_dead = -log_prob;

    return (alive > 0) ? loss_alive : loss_dead;
}

// Block reduction of 256 per-thread partials using V_WMMA_F32_16X16X4_F32.
// With B == all-ones, D[m][n] = sum_k A[m][k]: every A element contributes to
// every output column exactly once, so ANY bijection of 64 values into the
// A-matrix VGPRs yields the correct total after summing rows. Four
// accumulating WMMAs consume 256 values; then 8 VALU adds per lane and one
// lane^16 exchange finish the 16-row sum (D rows m=0..7 live in v[0..7] of
// lanes 0-15, m=8..15 in lanes 16-31).
__device__ __forceinline__ float block_reduce_wmma(float s, float* lds) {
    const int tid = threadIdx.x;
    lds[tid] = s;
    __syncthreads();
    float r = 0.0f;
    if (tid < 32) {                      // wave 0 only: branch is wave-uniform,
        v2f ones;                        // EXEC is all-ones inside (WMMA requirement)
        ones[0] = 1.0f; ones[1] = 1.0f;
        v8f acc = {};
#pragma unroll
        for (int j = 0; j < 4; ++j) {
            v2f a;
            a[0] = lds[tid + 64 * j];
            a[1] = lds[tid + 64 * j + 32];
            // 8 args: (neg_a, A, neg_b, B, c_mod, C, reuse_a, reuse_b)
            acc = __builtin_amdgcn_wmma_f32_16x16x4_f32(
                false, a, false, ones, (short)0, acc, false, false);
        }
        float s8 = acc[0] + acc[1] + acc[2] + acc[3] +
                   acc[4] + acc[5] + acc[6] + acc[7];
        r = s8 + __shfl_xor(s8, 16);     // combine lanes 0-15 with 16-31
    }
    return r;                            // valid on tid == 0
}

__global__ void __launch_bounds__(256)
loss_partial_kernel(const float4* __restrict__ pred,   // (N,2) -> N/2 float4
                    const float4* __restrict__ tte,    // N/4 float4
                    const int4*  __restrict__ alive,   // N/4 int4
                    float* __restrict__ partial,
                    int nv4)
{
    __shared__ float lds[256];
    const int tid = threadIdx.x;
    const int stride = gridDim.x * blockDim.x;
    float s = 0.0f;
    for (int g = blockIdx.x * blockDim.x + tid; g < nv4; g += stride) {
        float4 p0 = pred[2 * g];       // (mu0, ls0, mu1, ls1)
        float4 p1 = pred[2 * g + 1];   // (mu2, ls2, mu3, ls3)
        float4 t  = tte[g];
        int4   a  = alive[g];
        s += per_sample_loss(p0.x, p0.y, t.x, a.x);
        s += per_sample_loss(p0.z, p0.w, t.y, a.y);
        s += per_sample_loss(p1.x, p1.y, t.z, a.z);
        s += per_sample_loss(p1.z, p1.w, t.w, a.w);
    }
    float r = block_reduce_wmma(s, lds);
    if (tid == 0) partial[blockIdx.x] = r;
}

__global__ void __launch_bounds__(256)
final_reduce_kernel(const float* __restrict__ partial, int n_partials,
                    float* __restrict__ out, float inv_n)
{
    __shared__ float lds[256];
    const int tid = threadIdx.x;
    float s = 0.0f;
    for (int i = tid; i < n_partials; i += 256) s += partial[i];
    float r = block_reduce_wmma(s, lds);
    if (tid == 0) out[0] = r * inv_n;
}

extern "C" void kernel_launch(void* const* d_in, const int* in_sizes, int n_in,
                              void* d_out, int out_size, void* d_ws, size_t ws_size,
                              hipStream_t stream) {
    const float4* pred  = (const float4*)d_in[0];  // pred_params (N,2) f32
    const float4* tte   = (const float4*)d_in[1];  // tte (N,) f32
    const int4*   alive = (const int4*)d_in[2];    // is_alive (N,) i32
    const int n   = in_sizes[1];                   // N = 8388608 (multiple of 4)
    const int nv4 = n / 4;
    float* partial = (float*)d_ws;                 // 1024 floats of scratch

    const int BLOCKS = 1024;
    loss_partial_kernel<<<BLOCKS, 256, 0, stream>>>(pred, tte, alive, partial, nv4);
    final_reduce_kernel<<<1, 256, 0, stream>>>(partial, BLOCKS, (float*)d_out,
                                               1.0f / (float)n);
}